// Net_4518305596050
// MI455X (gfx1250) — compile-verified
//
#include <hip/hip_runtime.h>

// ---------------------------------------------------------------------------
// GraphConv(48 nodes, 8M edges) + MLP.
// Insight: segment_sum over 48 nodes == count-matrix C[48][48] (int) times x.
// Heavy work = streaming 128MB of int64 edge indices (bandwidth bound,
// ~5.5us at 23.3 TB/s). Dense math is tiny; C@x done with f32 WMMA
// (counts ~3500 exceed f16 exact-int range, so 16x16x4 F32 WMMA is correct).
// ---------------------------------------------------------------------------

typedef __attribute__((ext_vector_type(2))) float    v2f;
typedef __attribute__((ext_vector_type(8))) float    v8f;
typedef __attribute__((ext_vector_type(4))) unsigned u32x4;

#define NNODES 48
#define NBINS  (NNODES * NNODES)   // 2304

// ---------------- kernel 1: zero the global histogram (ws is poisoned) -----
__global__ void zero_hist(unsigned* __restrict__ hist) {
    int i = blockIdx.x * blockDim.x + threadIdx.x;
    if (i < NBINS) hist[i] = 0u;
}

// ---------------- kernel 2: edge histogram (the bandwidth-bound part) ------
// src_lo/dst_lo view the int64 arrays as u32 pairs; values < 48 so only the
// low dword matters. b128 NT loads grab 2 edges/lane/array per iteration.
__global__ __launch_bounds__(256) void edge_hist(
    const unsigned* __restrict__ src_lo,
    const unsigned* __restrict__ dst_lo,
    unsigned* __restrict__ ghist,
    long long n_edges)
{
    __shared__ unsigned lh[NBINS];
    for (int i = threadIdx.x; i < NBINS; i += blockDim.x) lh[i] = 0u;
    __syncthreads();

    long long n_pairs = n_edges >> 1;
    long long stride  = (long long)gridDim.x * blockDim.x;
    const u32x4* __restrict__ s128 = (const u32x4*)src_lo;
    const u32x4* __restrict__ d128 = (const u32x4*)dst_lo;

    for (long long p = (long long)blockIdx.x * blockDim.x + threadIdx.x;
         p < n_pairs; p += stride) {
        u32x4 s = __builtin_nontemporal_load(s128 + p);  // {lo,hi,lo,hi}
        u32x4 d = __builtin_nontemporal_load(d128 + p);
        atomicAdd(&lh[d.x * NNODES + s.x], 1u);          // ds_add_u32
        atomicAdd(&lh[d.z * NNODES + s.z], 1u);
    }
    // odd tail edge (n_edges is 8M in practice, but be safe)
    if ((n_edges & 1) && blockIdx.x == 0 && threadIdx.x == 0) {
        unsigned s = src_lo[2 * (n_edges - 1)];
        unsigned d = dst_lo[2 * (n_edges - 1)];
        atomicAdd(&lh[d * NNODES + s], 1u);
    }
    __syncthreads();
    for (int i = threadIdx.x; i < NBINS; i += blockDim.x) {
        unsigned v = lh[i];
        if (v) atomicAdd(&ghist[i], v);                  // global_atomic_add_u32
    }
}

// ---------------- kernel 3: all dense math in one block --------------------
__global__ __launch_bounds__(288) void dense_kernel(
    const unsigned* __restrict__ ghist,
    const float* __restrict__ x,
    const float* __restrict__ W_root1, const float* __restrict__ W_rel1,
    const float* __restrict__ b_rel1,
    const float* __restrict__ W_root2, const float* __restrict__ W_rel2,
    const float* __restrict__ b_rel2,
    const float* __restrict__ fc1_w, const float* __restrict__ fc1_b,
    const float* __restrict__ fc2_w, const float* __restrict__ fc2_b,
    float* __restrict__ out)
{
    __shared__ float C[NBINS];     // count matrix as f32 (exact: counts < 2^24)
    __shared__ float xs[288];      // x, row-major 48x6 == x.reshape(288)
    __shared__ float agg1[288];    // C @ x  (48x6)
    __shared__ float h[288];       // relu(conv1) (48x6)
    __shared__ float g[NNODES];    // h . W_rel2 per node
    __shared__ float v1[288];      // relu(fc1)

    const int tid = threadIdx.x;
    for (int i = tid; i < NBINS; i += blockDim.x) C[i] = (float)ghist[i];
    if (tid < 288) xs[tid] = x[tid];
    __syncthreads();

    // ---- agg1 = C(48x48) @ x(48x6) via V_WMMA_F32_16X16X4_F32 on wave 0 ----
    if (tid < 32) {
        const int lane = tid;
        const int grp  = lane >> 4;       // 0: K={0,1}; 1: K={2,3} per A layout
        const int l16  = lane & 15;

        // B fragments (x tile), built once, reused for all 3 row tiles.
        // B[k][n] = xs[k*6+n] for n<6 else 0; VGPR0 holds K=grp*2, VGPR1 K=grp*2+1.
        v2f Bf[12];
        for (int k4 = 0; k4 < 12; ++k4) {
            int k0 = k4 * 4 + grp * 2;
            float b0 = (l16 < 6) ? xs[k0 * 6 + l16]       : 0.f;
            float b1 = (l16 < 6) ? xs[(k0 + 1) * 6 + l16] : 0.f;
            v2f b = { b0, b1 };
            Bf[k4] = b;
        }
        for (int mt = 0; mt < 3; ++mt) {
            v8f acc = {};
            const int row = mt * 16 + l16;
            for (int k4 = 0; k4 < 12; ++k4) {
                int k0 = k4 * 4 + grp * 2;
                v2f A = { C[row * 48 + k0], C[row * 48 + k0 + 1] };
                acc = __builtin_amdgcn_wmma_f32_16x16x4_f32(
                          false, A, false, Bf[k4], (short)0, acc, false, false);
            }
            // D layout: VGPR r -> M=r (lanes 0-15) / M=r+8 (lanes 16-31), N=l16
            if (l16 < 6) {
                for (int r = 0; r < 8; ++r) {
                    int m = mt * 16 + r + grp * 8;
                    agg1[m * 6 + l16] = acc[r];
                }
            }
        }
    }
    __syncthreads();

    // ---- h = relu(agg1 @ W_rel1^T + b_rel1 + x @ W_root1^T), 288 threads ---
    {
        int m = tid / 6, n = tid % 6;
        float t = b_rel1[n];
        for (int f = 0; f < 6; ++f)
            t += agg1[m * 6 + f] * W_rel1[n * 6 + f]
               + xs[m * 6 + f]   * W_root1[n * 6 + f];
        h[tid] = fmaxf(t, 0.f);
    }
    __syncthreads();

    // ---- layer 2 folded: g[s] = h[s].W_rel2 ; h2[m] = relu(C[m,:].g + ...) -
    if (tid < NNODES) {
        float t = 0.f;
        for (int f = 0; f < 6; ++f) t += h[tid * 6 + f] * W_rel2[f];
        g[tid] = t;
    }
    __syncthreads();
    if (tid < NNODES) {
        float t = b_rel2[0];
        for (int s = 0; s < NNODES; ++s) t += C[tid * 48 + s] * g[s];
        for (int f = 0; f < 6; ++f)      t += h[tid * 6 + f] * W_root2[f];
        out[1 + tid] = fmaxf(t, 0.f);    // h2 occupies d_out[1..48]
    }

    // ---- MLP: v1 = relu(fc1_w @ xs + fc1_b); out = fc2_w . v1 + fc2_b -----
    {
        float t = fc1_b[tid];
        for (int j = 0; j < 288; ++j) t += fc1_w[tid * 288 + j] * xs[j];
        v1[tid] = fmaxf(t, 0.f);
    }
    __syncthreads();
    if (tid == 0) {
        float t = fc2_b[0];
        for (int j = 0; j < 288; ++j) t += fc2_w[j] * v1[j];
        out[0] = t;                      // scalar output at d_out[0]
    }
}

// ---------------------------------------------------------------------------
extern "C" void kernel_launch(void* const* d_in, const int* in_sizes, int n_in,
                              void* d_out, int out_size, void* d_ws, size_t ws_size,
                              hipStream_t stream) {
    const float*    x       = (const float*)d_in[0];
    const unsigned* ei      = (const unsigned*)d_in[1];   // int64 viewed as u32 pairs
    const float*    W_root1 = (const float*)d_in[2];
    const float*    W_rel1  = (const float*)d_in[3];
    const float*    b_rel1  = (const float*)d_in[4];
    const float*    W_root2 = (const float*)d_in[5];
    const float*    W_rel2  = (const float*)d_in[6];
    const float*    b_rel2  = (const float*)d_in[7];
    const float*    fc1_w   = (const float*)d_in[8];
    const float*    fc1_b   = (const float*)d_in[9];
    const float*    fc2_w   = (const float*)d_in[10];
    const float*    fc2_b   = (const float*)d_in[11];

    long long n_edges = (long long)in_sizes[1] / 2;       // edge_index is (2, E)
    const unsigned* src_lo = ei;                          // row 0: src
    const unsigned* dst_lo = ei + 2 * n_edges;            // row 1: dst (int64 -> 2 dwords)

    unsigned* hist = (unsigned*)d_ws;                     // 2304 * 4B = 9216B scratch

    zero_hist<<<(NBINS + 255) / 256, 256, 0, stream>>>(hist);
    edge_hist<<<512, 256, 0, stream>>>(src_lo, dst_lo, hist, n_edges);
    dense_kernel<<<1, 288, 0, stream>>>(hist, x,
                                        W_root1, W_rel1, b_rel1,
                                        W_root2, W_rel2, b_rel2,
                                        fc1_w, fc1_b, fc2_w, fc2_b,
                                        (float*)d_out);
}